// SparseTopKLayer_75041668596072
// MI455X (gfx1250) — compile-verified
//
#include <hip/hip_runtime.h>

// SparseTopKLayer for MI455X (gfx1250, wave32).
// N=32768 rows, D=2048 cols fp32; k read from device pointer.
// One workgroup per row; 256 threads x 8 elements, fully register-resident.
// Memory-bound: 1x NT read of x, 1x NT write of out (512 MB -> ~22us @ 23.3 TB/s);
// weight/gamma stay regular-temporal so they live in the 192MB L2 across blocks.

#define D_COLS   2048
#define TPB      256
#define EPS_F    1e-6f

typedef float v4f __attribute__((ext_vector_type(4)));

__global__ __launch_bounds__(TPB) void sparse_topk_rmsnorm_kernel(
    const float* __restrict__ x,
    const float* __restrict__ weight,
    const float* __restrict__ gamma,
    const int*   __restrict__ kptr,
    float* __restrict__ out)
{
    __shared__ unsigned s_hist[256];   // radix histogram
    __shared__ float    s_red[8];      // per-wave sumsq partials
    __shared__ unsigned s_bin;         // selected digit (broadcast)
    __shared__ unsigned s_knext;       // residual k (broadcast)

    const int tid  = threadIdx.x;
    const long base = (long)blockIdx.x * D_COLS;

    // ---- coalesced 128-bit loads; x is streamed non-temporal (>> L2 capacity) ----
    const v4f* xr = (const v4f*)(x + base);
    v4f a = __builtin_nontemporal_load(xr + tid);        // elements 4*tid .. 4*tid+3
    v4f b = __builtin_nontemporal_load(xr + TPB + tid);  // elements 1024+4*tid ..

    const v4f* wv = (const v4f*)weight;                  // 8KB, reused by all blocks: RT
    const v4f* gv = (const v4f*)gamma;
    v4f w0 = wv[tid];
    v4f w1 = wv[TPB + tid];
    v4f g0 = gv[tid];
    v4f g1 = gv[TPB + tid];

    float xs[8] = {a.x, a.y, a.z, a.w, b.x, b.y, b.z, b.w};
    float ws[8] = {w0.x, w0.y, w0.z, w0.w, w1.x, w1.y, w1.z, w1.w};
    float gs[8] = {g0.x, g0.y, g0.z, g0.w, g1.x, g1.y, g1.z, g1.w};

    // ---- sum of squares: wave32 shfl reduce, then cross-wave via LDS ----
    float ss = 0.0f;
#pragma unroll
    for (int i = 0; i < 8; ++i) ss += xs[i] * xs[i];
#pragma unroll
    for (int off = 16; off > 0; off >>= 1)
        ss += __shfl_xor(ss, off, 32);
    if ((tid & 31) == 0) s_red[tid >> 5] = ss;
    __syncthreads();
    float tot = 0.0f;
#pragma unroll
    for (int i = 0; i < 8; ++i) tot += s_red[i];
    const float rms = rsqrtf(tot * (1.0f / (float)D_COLS) + EPS_F);

    // ---- normalized values and magnitude bit patterns (non-negative IEEE:
    //      uint ordering == float ordering) ----
    float    xn[8];
    unsigned u[8];
#pragma unroll
    for (int i = 0; i < 8; ++i) {
        xn[i] = xs[i] * rms * ws[i];
        u[i]  = __float_as_uint(fabsf(xn[i]));
    }

    // ---- exact radix select of k-th largest over 2048 register-resident values ----
    unsigned kcur   = (unsigned)kptr[0];   // k = 256
    unsigned prefix = 0u;

#pragma unroll
    for (int shift = 24; shift >= 0; shift -= 8) {
        // clear histogram (256 bins, one per thread).
        // NOTE: no extra barrier needed before this — the previous pass's s_bin
        // read is fenced by the two barriers below before s_bin is rewritten.
        s_hist[tid] = 0u;
        __syncthreads();

        // candidates: high bits above this digit must match prefix
        const unsigned himask = (shift == 24) ? 0u : (0xFFFFFFFFu << (shift + 8));
#pragma unroll
        for (int i = 0; i < 8; ++i) {
            if ((u[i] & himask) == prefix)
                atomicAdd(&s_hist[(u[i] >> shift) & 0xFFu], 1u);
        }
        __syncthreads();

        // single wave scans 256 bins from the top (descending) to find the digit
        if (tid < 32) {
            const int lane = tid;
            unsigned c[8];
            unsigned gsum = 0u;
#pragma unroll
            for (int j = 0; j < 8; ++j) {
                c[j] = s_hist[255 - lane * 8 - j];   // descending bin order
                gsum += c[j];
            }
            // inclusive scan over lanes (lane 0 owns the highest bins)
            unsigned pre = gsum;
#pragma unroll
            for (int off = 1; off < 32; off <<= 1) {
                unsigned o = __shfl_up(pre, off, 32);
                if (lane >= off) pre += o;
            }
            pre -= gsum;                              // exclusive: count strictly above my group
            if (pre < kcur && kcur <= pre + gsum) {   // exactly one lane matches
                unsigned acc = pre;
#pragma unroll
                for (int j = 0; j < 8; ++j) {
                    unsigned nacc = acc + c[j];
                    if (nacc >= kcur) {
                        s_bin   = (unsigned)(255 - lane * 8 - j);
                        s_knext = kcur - acc;         // residual rank within this bin
                        break;
                    }
                    acc = nacc;
                }
            }
        }
        __syncthreads();
        prefix |= (s_bin << shift);
        kcur    = s_knext;
    }

    const unsigned thr = prefix;   // exact bit pattern of k-th largest magnitude

    // ---- masked residual add, coalesced non-temporal 128-bit stores ----
    float o[8];
#pragma unroll
    for (int i = 0; i < 8; ++i)
        o[i] = xs[i] + ((u[i] >= thr) ? xn[i] * gs[i] : 0.0f);

    v4f* orow = (v4f*)(out + base);
    v4f o0 = {o[0], o[1], o[2], o[3]};
    v4f o1 = {o[4], o[5], o[6], o[7]};
    __builtin_nontemporal_store(o0, orow + tid);
    __builtin_nontemporal_store(o1, orow + TPB + tid);
}

extern "C" void kernel_launch(void* const* d_in, const int* in_sizes, int n_in,
                              void* d_out, int out_size, void* d_ws, size_t ws_size,
                              hipStream_t stream) {
    const float* x      = (const float*)d_in[0];
    const float* weight = (const float*)d_in[1];
    const float* gamma  = (const float*)d_in[2];
    const int*   k      = (const int*)d_in[3];
    float*       out    = (float*)d_out;

    const int n_rows = in_sizes[0] / D_COLS;   // 32768
    sparse_topk_rmsnorm_kernel<<<n_rows, TPB, 0, stream>>>(x, weight, gamma, k, out);
}